// GNN_21191368638621
// MI455X (gfx1250) — compile-verified
//
#include <hip/hip_runtime.h>
#include <hip/hip_bf16.h>

typedef float v2f __attribute__((ext_vector_type(2)));
typedef float v4f __attribute__((ext_vector_type(4)));
typedef float v8f __attribute__((ext_vector_type(8)));
typedef int   v4i __attribute__((ext_vector_type(4)));

// ---------------------------------------------------------------------------
// Edge pass, 4 edges per thread via 128-bit non-temporal streaming loads.
// WITH_DEG: agg is interleaved [agg, deg] pairs -> both atomics per edge hit
// the same L2 sector. Otherwise plain scalar scatter (layer-2 projected msg).
// node_val (2MB) and the atomic targets (2-4MB) stay resident in the 192MB
// L2; the NT hints keep the 384MB of streamed src/dst/e_feat from rinsing it.
// ---------------------------------------------------------------------------
template <bool WITH_DEG>
__global__ void gnn_edge_scatter(const float* __restrict__ node_val,
                                 const float* __restrict__ e_feat,
                                 const int* __restrict__ src,
                                 const int* __restrict__ dst,
                                 float* __restrict__ agg,
                                 int n_edges)
{
    const int base = (blockIdx.x * blockDim.x + threadIdx.x) * 4;
    if (base + 3 < n_edges) {
        v4i s4 = __builtin_nontemporal_load((const v4i*)(src + base));
        v4i d4 = __builtin_nontemporal_load((const v4i*)(dst + base));
        v4f w4 = __builtin_nontemporal_load((const v4f*)(e_feat + base));
        #pragma unroll
        for (int k = 0; k < 4; ++k) {
            float msg = node_val[s4[k]] * w4[k];
            if (WITH_DEG) {
                atomicAdd(&agg[2 * d4[k] + 0], msg);
                atomicAdd(&agg[2 * d4[k] + 1], 1.0f);
            } else {
                atomicAdd(&agg[d4[k]], msg);
            }
        }
    } else if (base < n_edges) {
        for (int i = base; i < n_edges; ++i) {
            float msg = node_val[src[i]] * e_feat[i];
            if (WITH_DEG) {
                atomicAdd(&agg[2 * dst[i] + 0], msg);
                atomicAdd(&agg[2 * dst[i] + 1], 1.0f);
            } else {
                atomicAdd(&agg[dst[i]], msg);
            }
        }
    }
}

// ---------------------------------------------------------------------------
// Node transform, one wave32 handles 16 nodes via V_WMMA_F32_16X16X4_F32:
//   A (16x4):  row m = [ x_m, h_neigh_m, 1, 0 ]
//   B (4x16):  col j = [ W1_self[j], W1_neigh[j], b1[j], 0 ]  (j>=10 -> 0)
//   D (16x16): pre-activation of layer 1 (bias folded in via K=2 ones col)
// Sigmoid (v_exp + v_rcp) on D, then the two layer-2 projections as 10-wide
// dot products reduced across the 16 N-lanes (xor butterfly):
//   s_out[m] = h[m] @ W2_neigh      (scalar scatter source for layer 2)
//   oself[m] = h[m] @ W2_self + b2  (self term of layer 2)
// h[N,10] is never materialized.
// ---------------------------------------------------------------------------
__global__ void gnn_node_wmma(const float* __restrict__ in_feat,
                              const float* __restrict__ agg_deg,   // [N][2]
                              const float* __restrict__ W1_self,
                              const float* __restrict__ W1_neigh,
                              const float* __restrict__ b1,
                              const float* __restrict__ W2_self,
                              const float* __restrict__ W2_neigh,
                              const float* __restrict__ b2,
                              float* __restrict__ s_out,
                              float* __restrict__ oself,
                              int n_nodes)
{
    const int lane = threadIdx.x & 31;
    const int wave = (blockIdx.x * blockDim.x + threadIdx.x) >> 5;
    const int base = wave * 16;                 // 16 nodes per wave
    if (base >= n_nodes) return;                // wave-uniform exit

    const int j  = lane & 15;                   // N index (output channel)
    const int hi = lane >> 4;                   // 0: K=0,1   1: K=2,3

    // --- gather per-node inputs (lanes 16-31 redundantly load same addrs) ---
    int m = base + j; if (m >= n_nodes) m = n_nodes - 1;
    float x  = in_feat[m];
    v2f  ad  = *(const v2f*)(agg_deg + 2 * (size_t)m);   // {agg1, deg}
    float dg = ad.y;
    float hn = (dg > 0.0f) ? ad.x * __builtin_amdgcn_rcpf(dg) : 0.0f;

    // --- per-channel weights (zero-padded for j >= 10) ---
    float w1s = (j < 10) ? W1_self[j]  : 0.0f;
    float w1n = (j < 10) ? W1_neigh[j] : 0.0f;
    float bb1 = (j < 10) ? b1[j]       : 0.0f;
    float w2s = (j < 10) ? W2_self[j]  : 0.0f;
    float w2n = (j < 10) ? W2_neigh[j] : 0.0f;
    float bb2 = b2[0];

    // --- build A and B per the 16x16x4 f32 VGPR layout (ISA §7.12.2) ---
    v2f a;                                      // A: node rows
    a.x = hi ? 1.0f : x;                        // K=0 (x) | K=2 (ones -> bias)
    a.y = hi ? 0.0f : hn;                       // K=1 (h_neigh) | K=3 (zero)
    v2f b;                                      // B: weight rows
    b.x = hi ? bb1  : w1s;                      // K=0 W_self | K=2 b1
    b.y = hi ? 0.0f : w1n;                      // K=1 W_neigh | K=3 zero

    v8f d;
#if __has_builtin(__builtin_amdgcn_wmma_f32_16x16x4_f32)
    v8f c = {0.f, 0.f, 0.f, 0.f, 0.f, 0.f, 0.f, 0.f};
    d = __builtin_amdgcn_wmma_f32_16x16x4_f32(
            /*neg_a=*/false, a, /*neg_b=*/false, b,
            /*c_mod=*/(short)0, c, /*reuse_a=*/false, /*reuse_b=*/false);
#else
    #pragma unroll
    for (int v = 0; v < 8; ++v) {               // VALU fallback, same D layout
        int mrow = v + 8 * hi;
        float xm  = __shfl(x,  mrow, 32);
        float hnm = __shfl(hn, mrow, 32);
        d[v] = xm * w1s + hnm * w1n + bb1;
    }
#endif

    // --- sigmoid (fast rcp) + both projections, reduced over 16 N-lanes ---
    float sv[8], ov[8];
    #pragma unroll
    for (int v = 0; v < 8; ++v) {
        float hval = __builtin_amdgcn_rcpf(1.0f + __expf(-d[v]));
        hval = (j < 10) ? hval : 0.0f;
        float ssum = hval * w2n;
        float osum = hval * w2s;
        #pragma unroll
        for (int off = 8; off >= 1; off >>= 1) {
            ssum += __shfl_xor(ssum, off, 32);  // stays inside each 16-lane half
            osum += __shfl_xor(osum, off, 32);
        }
        sv[v] = ssum;
        ov[v] = osum;
    }

    // lane j==v in each half writes node (base + 8*hi + v)
    #pragma unroll
    for (int v = 0; v < 8; ++v) {
        if (j == v) {
            int mo = base + 8 * hi + v;
            if (mo < n_nodes) {
                s_out[mo] = sv[v];
                oself[mo] = ov[v] + bb2;
            }
        }
    }
}

// ---------------------------------------------------------------------------
// out[n] = oself[n] + (deg > 0 ? agg2s[n] / deg[n] : 0)
// ---------------------------------------------------------------------------
__global__ void gnn_finalize(const float* __restrict__ oself,
                             const float* __restrict__ agg2s,
                             const float* __restrict__ agg_deg,  // [N][2]
                             float* __restrict__ out,
                             int n_nodes)
{
    int i = blockIdx.x * blockDim.x + threadIdx.x;
    if (i >= n_nodes) return;
    float dg = agg_deg[2 * (size_t)i + 1];
    float neigh = (dg > 0.0f) ? agg2s[i] * __builtin_amdgcn_rcpf(dg) : 0.0f;
    out[i] = oself[i] + neigh;
}

extern "C" void kernel_launch(void* const* d_in, const int* in_sizes, int n_in,
                              void* d_out, int out_size, void* d_ws, size_t ws_size,
                              hipStream_t stream) {
    const float* in_feat  = (const float*)d_in[0];   // [N,1]
    const float* e_feat   = (const float*)d_in[1];   // [E,1]
    const int*   src      = (const int*)  d_in[2];   // [E]
    const int*   dst      = (const int*)  d_in[3];   // [E]
    const float* W1_self  = (const float*)d_in[4];   // [1,10]
    const float* W1_neigh = (const float*)d_in[5];   // [1,10]
    const float* b1       = (const float*)d_in[6];   // [10]
    const float* W2_self  = (const float*)d_in[7];   // [10,1]
    const float* W2_neigh = (const float*)d_in[8];   // [10,1]
    const float* b2       = (const float*)d_in[9];   // [1]
    float*       out      = (float*)d_out;

    const int N = in_sizes[0];                       // 500,000 nodes
    const int E = in_sizes[2];                       // 16,000,000 edges

    // Workspace: [agg_deg interleaved (2N) | agg2s (N) | s (N) | oself (N)]
    float* ws      = (float*)d_ws;
    float* agg_deg = ws;                             // [N][2] = {agg1, deg}
    float* agg2s   = ws + (size_t)2 * N;
    float* s_prj   = ws + (size_t)3 * N;
    float* oself   = ws + (size_t)4 * N;

    // zero the accumulators (agg_deg + agg2s are contiguous: 3N floats)
    hipMemsetAsync(ws, 0, (size_t)3 * N * sizeof(float), stream);

    const int BLK = 256;
    const int eThreads = (E + 3) / 4;                // 4 edges per thread
    const int eBlocks  = (eThreads + BLK - 1) / BLK;
    const int waves    = (N + 15) / 16;              // 16 nodes per wave32
    const int nBlocks  = (waves * 32 + BLK - 1) / BLK;
    const int fBlocks  = (N + BLK - 1) / BLK;

    // Layer 1 edge pass: agg_deg[dst] += {in_feat[src]*w, 1}
    gnn_edge_scatter<true><<<eBlocks, BLK, 0, stream>>>(in_feat, e_feat, src,
                                                        dst, agg_deg, E);
    // Fused node transform (WMMA) -> s_prj, oself
    gnn_node_wmma<<<nBlocks, BLK, 0, stream>>>(in_feat, agg_deg,
                                               W1_self, W1_neigh, b1,
                                               W2_self, W2_neigh, b2,
                                               s_prj, oself, N);
    // Layer 2 edge pass (projection commutes with mean): scalar scatter
    gnn_edge_scatter<false><<<eBlocks, BLK, 0, stream>>>(s_prj, e_feat, src,
                                                         dst, agg2s, E);
    // out = oself + agg2s/deg
    gnn_finalize<<<fBlocks, BLK, 0, stream>>>(oself, agg2s, agg_deg, out, N);
}